// Cam2BEV_32427003084903
// MI455X (gfx1250) — compile-verified
//
#include <hip/hip_runtime.h>
#include <hip/hip_bf16.h>
#include <math.h>

typedef __attribute__((ext_vector_type(2))) float v2f;
typedef __attribute__((ext_vector_type(8))) float v8f;

// ---------------- problem constants ----------------
#define BB 1
#define NCAM 4
#define FC 256
#define HH 45
#define WW 80
#define PIX (HH * WW)          // 3600
#define ORH 360
#define ORW 640
#define DD 99
#define CC 128
#define OC3 (DD + CC)          // 227
#define NX 100
#define NY 100
#define NZ 5
#define NPTS (NCAM * DD * PIX) // 1,425,600
#define OUT_ELEMS (NZ * CC * NX * NY) // 6,400,000

// ---------------- small helpers ----------------
__device__ inline void invert3(const float* a, float* o) {
    float det = a[0]*(a[4]*a[8]-a[5]*a[7])
              - a[1]*(a[3]*a[8]-a[5]*a[6])
              + a[2]*(a[3]*a[7]-a[4]*a[6]);
    float id = 1.0f / det;
    o[0]=(a[4]*a[8]-a[5]*a[7])*id; o[1]=(a[2]*a[7]-a[1]*a[8])*id; o[2]=(a[1]*a[5]-a[2]*a[4])*id;
    o[3]=(a[5]*a[6]-a[3]*a[8])*id; o[4]=(a[0]*a[8]-a[2]*a[6])*id; o[5]=(a[2]*a[3]-a[0]*a[5])*id;
    o[6]=(a[3]*a[7]-a[4]*a[6])*id; o[7]=(a[1]*a[6]-a[0]*a[7])*id; o[8]=(a[0]*a[4]-a[1]*a[3])*id;
}

// CDNA5 async global->LDS copy (ASYNCcnt-tracked, bypasses VGPRs).
// vdst = LDS byte address (wave-relative, low 32 bits of the flat shared ptr),
// vaddr = 64-bit global address, GV addressing mode.
__device__ inline void async_g2l_b32(unsigned lds_off, const float* gptr) {
    unsigned long long ga = (unsigned long long)(const void*)gptr;
    asm volatile("global_load_async_to_lds_b32 %0, %1, off"
                 :: "v"(lds_off), "v"(ga) : "memory");
}
__device__ inline void wait_asynccnt0() {
    asm volatile("s_wait_asynccnt 0x0" ::: "memory");
}

// ---------------- per-camera matrix precompute ----------------
// cam[n*24 + 0..8]  = inv(post_rot)
// cam[n*24 + 9..17] = cam2ego_rot @ inv(K)
// cam[n*24 +18..20] = cam2ego_trans
// cam[n*24 +21..23] = post_trans
__global__ void precam_kernel(const float* __restrict__ c2e_rot,
                              const float* __restrict__ c2e_trans,
                              const float* __restrict__ intr,
                              const float* __restrict__ post_rot,
                              const float* __restrict__ post_trans,
                              float* __restrict__ cam) {
    int n = threadIdx.x;
    if (n >= NCAM) return;
    float prInv[9], kInv[9], comb[9];
    invert3(post_rot + n * 9, prInv);
    invert3(intr + n * 9, kInv);
    const float* R = c2e_rot + n * 9;
    for (int i = 0; i < 3; ++i)
        for (int j = 0; j < 3; ++j)
            comb[i*3+j] = R[i*3+0]*kInv[0*3+j] + R[i*3+1]*kInv[1*3+j] + R[i*3+2]*kInv[2*3+j];
    float* o = cam + n * 24;
    for (int i = 0; i < 9; ++i) { o[i] = prInv[i]; o[9+i] = comb[i]; }
    for (int i = 0; i < 3; ++i) { o[18+i] = c2e_trans[n*3+i]; o[21+i] = post_trans[n*3+i]; }
}

// ---------------- geometry -> output-base index ----------------
// segbase[(n*DD + d)*PIX + p] = z*C*NX*NY + x*NY + y  (or -1 if outside grid)
__global__ __launch_bounds__(256) void geom_kernel(const float* __restrict__ cam,
                                                   int* __restrict__ segbase) {
    int idx = blockIdx.x * 256 + threadIdx.x;
    if (idx >= NPTS) return;
    int p = idx % PIX;
    int t = idx / PIX;
    int d = t % DD;
    int n = t / DD;
    int h = p / WW, w = p % WW;

    float xs = (float)w * ((float)(ORW - 1) / (float)(WW - 1));
    float ys = (float)h * ((float)(ORH - 1) / (float)(HH - 1));
    float zs = 1.0f + (float)d;

    const float* c = cam + n * 24;
    // undo post transform
    float x0 = xs - c[21], y0 = ys - c[22], z0 = zs - c[23];
    float x1 = c[0]*x0 + c[1]*y0 + c[2]*z0;
    float y1 = c[3]*x0 + c[4]*y0 + c[5]*z0;
    float z1 = c[6]*x0 + c[7]*y0 + c[8]*z0;
    // un-project: (u*z, v*z, z)
    float px = x1 * z1, py = y1 * z1, pz = z1;
    // camera -> ego
    float gx = c[ 9]*px + c[10]*py + c[11]*pz + c[18];
    float gy = c[12]*px + c[13]*py + c[14]*pz + c[19];
    float gz = c[15]*px + c[16]*py + c[17]*pz + c[20];
    // voxelize: lower bound = BXV - DX/2 = (0, -100, -10); DX = (2,2,4)
    int vx = (int)floorf(gx * 0.5f);
    int vy = (int)floorf((gy + 100.0f) * 0.5f);
    int vz = (int)floorf((gz + 10.0f) * 0.25f);
    int base = -1;
    if (vx >= 0 && vx < NX && vy >= 0 && vy < NY && vz >= 0 && vz < NZ)
        base = vz * (CC * NX * NY) + vx * NY + vy;
    segbase[idx] = base;
}

// ---------------- zero the output ----------------
__global__ __launch_bounds__(256) void zero_kernel(float* __restrict__ out, int n) {
    int i = blockIdx.x * 256 + threadIdx.x;
    if (i < n) out[i] = 0.0f;
}

// ---------------- implicit-GEMM conv via V_WMMA_F32_16X16X4_F32 ----------------
// in : (img, IC, H, W) fp32        wgt: (OC, IC, kdim, kdim)
// out: (img, OC, H, W)             optional BN (+1e-3 eps) and ReLU fused
// block = 128 threads (4 wave32); block tile = 64 pixels x 64 out-channels
// wave tile = 16 pixels x 64 channels (4 x v8f accumulators)
// Staging: double-buffered LDS filled by CDNA5 async global->LDS copies,
// overlapped with the WMMA work of the current stage.
__global__ __launch_bounds__(128) void conv_wmma_kernel(
        const float* __restrict__ in, const float* __restrict__ wgt,
        const float* __restrict__ bias,
        const float* __restrict__ gamma, const float* __restrict__ beta,
        const float* __restrict__ mean, const float* __restrict__ var,
        float* __restrict__ out,
        int IC, int OC, int kdim, int bnrelu) {
    __shared__ float lds_a[2][32 * 64];   // [buf][k][m]  A = activations (pixels x K)
    __shared__ float lds_b[2][32 * 64];   // [buf][k][n]  B = weights (K x out-channels)

    const int tid  = threadIdx.x;
    const int lane = tid & 31;
    const int wave = tid >> 5;
    const int mTile = blockIdx.x * 64;   // pixel base
    const int nTile = blockIdx.y * 64;   // out-channel base
    const int img   = blockIdx.z;

    const float* inImg = in + (size_t)img * IC * PIX;
    const int taps = kdim * kdim;
    const int kr = kdim >> 1;
    const int chunks = IC >> 5;          // IC / 32
    const int nStages = taps * chunks;

    v8f acc[4] = {};

    const int mLane = lane & 15;           // pixel-in-strip / channel-in-subtile
    const int kHalf = (lane >> 4) << 1;    // 0 for lanes 0-15, 2 for lanes 16-31

    // issue async fills for stage `st` into buffer `buf`
    auto issue_stage = [&](int st, int buf) {
        int t   = st / chunks;
        int ic0 = (st - t * chunks) << 5;
        int dy  = t / kdim - kr;
        int dx  = t % kdim - kr;
        // ---- A tile: lds_a[buf][k][m] = in[ic0+k][h+dy][w+dx] (0 on halo/tail) ----
        for (int idx = tid; idx < 32 * 64; idx += 128) {
            int k = idx >> 6, mm = idx & 63;
            int p = mTile + mm;
            int h = p / WW + dy, w = p % WW + dx;
            bool ok = (p < PIX) && (h >= 0) && (h < HH) && (w >= 0) && (w < WW);
            if (ok) {
                async_g2l_b32((unsigned)(size_t)&lds_a[buf][k * 64 + mm],
                              inImg + (size_t)(ic0 + k) * PIX + h * WW + w);
            } else {
                lds_a[buf][k * 64 + mm] = 0.0f;   // zero-pad via DS store
            }
        }
        // ---- B tile: lds_b[buf][k][n] = wgt[oc][ic0+k][tap] ----
        for (int idx = tid; idx < 32 * 64; idx += 128) {
            int k = idx >> 6, nn = idx & 63;
            int oc = nTile + nn;
            if (oc >= OC) oc = OC - 1;   // clamp: those columns are never stored
            async_g2l_b32((unsigned)(size_t)&lds_b[buf][k * 64 + nn],
                          wgt + ((size_t)oc * IC + (ic0 + k)) * taps + t);
        }
    };

    // prologue: fill buffer 0
    issue_stage(0, 0);
    wait_asynccnt0();
    __syncthreads();

    for (int st = 0; st < nStages; ++st) {
        const int cur = st & 1;
        // overlap: start filling the other buffer while we compute this one
        if (st + 1 < nStages) issue_stage(st + 1, cur ^ 1);

        const float* la = lds_a[cur];
        const float* lb = lds_b[cur];
        // ---- 8 WMMA K-steps of 4 ----
        #pragma unroll
        for (int kk = 0; kk < 32; kk += 4) {
            int k0 = kk + kHalf;             // ISA A layout: VGPR0=K+0/K+2, VGPR1=K+1/K+3
            v2f a;
            a.x = la[k0 * 64 + wave * 16 + mLane];
            a.y = la[(k0 + 1) * 64 + wave * 16 + mLane];
            #pragma unroll
            for (int s = 0; s < 4; ++s) {
                v2f b;
                b.x = lb[k0 * 64 + s * 16 + mLane];
                b.y = lb[(k0 + 1) * 64 + s * 16 + mLane];
                acc[s] = __builtin_amdgcn_wmma_f32_16x16x4_f32(
                    false, a, false, b, (short)0, acc[s], false, false);
            }
        }
        // make next buffer's async fills visible to all waves
        wait_asynccnt0();
        __syncthreads();
    }

    // ---- epilogue: bias (+ BN + ReLU), store NCHW ----
    #pragma unroll
    for (int s = 0; s < 4; ++s) {
        int oc = nTile + s * 16 + mLane;
        if (oc >= OC) continue;
        float bi = bias[oc];
        float scale = 1.0f, shift = 0.0f;
        if (bnrelu) {
            float inv = rsqrtf(var[oc] + 1e-3f);
            scale = gamma[oc] * inv;
            shift = beta[oc] - mean[oc] * scale;
        }
        float* outCh = out + ((size_t)img * OC + oc) * PIX;
        #pragma unroll
        for (int r = 0; r < 8; ++r) {
            int mm = r + ((lane >> 4) << 3);     // ISA C layout: VGPR r = M=r / M=r+8
            int p = mTile + wave * 16 + mm;
            if (p < PIX) {
                float val = acc[s][r] + bi;
                if (bnrelu) { val = val * scale + shift; val = fmaxf(val, 0.0f); }
                outCh[p] = val;
            }
        }
    }
}

// ---------------- softmax over depth + outer-product + voxel scatter ----------------
// one block of 128 threads per (camera, pixel); thread c owns context channel c
__global__ __launch_bounds__(128) void splat_kernel(const float* __restrict__ x3,
                                                    const int* __restrict__ segbase,
                                                    float* __restrict__ out) {
    __shared__ float sm[128];
    __shared__ float depth[DD];

    const int p   = blockIdx.x;   // 0..3599
    const int n   = blockIdx.y;   // 0..3
    const int tid = threadIdx.x;  // 0..127

    // depth logits -> softmax (99 valid entries)
    float v = (tid < DD) ? x3[((size_t)n * OC3 + tid) * PIX + p] : -INFINITY;
    sm[tid] = v;
    __syncthreads();
    #pragma unroll
    for (int s = 64; s > 0; s >>= 1) {
        if (tid < s) sm[tid] = fmaxf(sm[tid], sm[tid + s]);
        __syncthreads();
    }
    float mx = sm[0];
    __syncthreads();
    float e = (tid < DD) ? __expf(v - mx) : 0.0f;
    sm[tid] = e;
    __syncthreads();
    #pragma unroll
    for (int s = 64; s > 0; s >>= 1) {
        if (tid < s) sm[tid] += sm[tid + s];
        __syncthreads();
    }
    float invSum = 1.0f / sm[0];
    if (tid < DD) depth[tid] = e * invSum;

    // context channel for this thread
    float ctx = x3[((size_t)n * OC3 + DD + tid) * PIX + p];
    __syncthreads();

    const int* sb = segbase + (size_t)n * DD * PIX + p;
    #pragma unroll 4
    for (int d = 0; d < DD; ++d) {
        int base = sb[(size_t)d * PIX];
        if (base >= 0)
            atomicAdd(out + base + tid * (NX * NY), depth[d] * ctx);
    }
}

// ---------------- host-side launcher ----------------
extern "C" void kernel_launch(void* const* d_in, const int* in_sizes, int n_in,
                              void* d_out, int out_size, void* d_ws, size_t ws_size,
                              hipStream_t stream) {
    const float* c2e_rot    = (const float*)d_in[0];
    const float* c2e_trans  = (const float*)d_in[1];
    const float* intr       = (const float*)d_in[2];
    const float* post_rot   = (const float*)d_in[3];
    const float* post_trans = (const float*)d_in[4];
    const float* img_feats  = (const float*)d_in[5];
    const float* w1 = (const float*)d_in[6];
    const float* b1 = (const float*)d_in[7];
    const float* g1 = (const float*)d_in[8];
    const float* be1= (const float*)d_in[9];
    const float* m1 = (const float*)d_in[10];
    const float* v1 = (const float*)d_in[11];
    const float* w2 = (const float*)d_in[12];
    const float* b2 = (const float*)d_in[13];
    const float* g2 = (const float*)d_in[14];
    const float* be2= (const float*)d_in[15];
    const float* m2 = (const float*)d_in[16];
    const float* v2 = (const float*)d_in[17];
    const float* w3 = (const float*)d_in[18];
    const float* b3 = (const float*)d_in[19];
    float* out = (float*)d_out;

    // workspace layout (all 256B aligned)
    char* ws = (char*)d_ws;
    size_t off = 0;
    auto alloc = [&](size_t bytes) { char* p = ws + off; off = (off + bytes + 255) & ~(size_t)255; return p; };
    float* cam  = (float*)alloc(NCAM * 24 * sizeof(float));
    int*   segb = (int*)  alloc((size_t)NPTS * sizeof(int));
    float* act1 = (float*)alloc((size_t)NCAM * FC  * PIX * sizeof(float));
    float* act2 = (float*)alloc((size_t)NCAM * FC  * PIX * sizeof(float));
    float* x3   = (float*)alloc((size_t)NCAM * OC3 * PIX * sizeof(float));
    (void)ws_size; (void)in_sizes; (void)n_in; (void)out_size;

    // 1) per-camera matrices
    precam_kernel<<<1, 32, 0, stream>>>(c2e_rot, c2e_trans, intr, post_rot, post_trans, cam);

    // 2) geometry -> voxel base indices
    geom_kernel<<<(NPTS + 255) / 256, 256, 0, stream>>>(cam, segb);

    // 3) zero BEV output (harness poisons it)
    zero_kernel<<<(OUT_ELEMS + 255) / 256, 256, 0, stream>>>(out, OUT_ELEMS);

    // 4-6) conv stack (implicit GEMM on fp32 WMMA, async-LDS double buffering)
    dim3 cgrid((PIX + 63) / 64, (FC + 63) / 64, NCAM);
    conv_wmma_kernel<<<cgrid, 128, 0, stream>>>(img_feats, w1, b1, g1, be1, m1, v1,
                                                act1, FC, FC, 3, 1);
    conv_wmma_kernel<<<cgrid, 128, 0, stream>>>(act1, w2, b2, g2, be2, m2, v2,
                                                act2, FC, FC, 3, 1);
    dim3 cgrid3((PIX + 63) / 64, (OC3 + 63) / 64, NCAM);
    conv_wmma_kernel<<<cgrid3, 128, 0, stream>>>(act2, w3, b3, nullptr, nullptr, nullptr, nullptr,
                                                 x3, FC, OC3, 1, 0);

    // 7) softmax + outer product + scatter into BEV
    dim3 sgrid(PIX, NCAM);
    splat_kernel<<<sgrid, 128, 0, stream>>>(x3, segb, out);
}